// ScaledDotProductAttention_69741678952654
// MI455X (gfx1250) — compile-verified
//
#include <hip/hip_runtime.h>

// Scaled dot-product attention, B=1 H=16 L=4096 D=64, fp32.
// Outputs: out [B,H,L,D] then score [B,H,L,L], concatenated in d_out.
//
// Fused design: one workgroup = one head x one 16-row query block.
//  - 16x4096 f32 score slab lives in LDS (CDNA5 WGP has 320KB LDS).
//  - Pass 1: S = (Q/sqrt(D)) K^T via v_wmma_f32_16x16x4_f32, slab -> LDS.
//  - Pass 2: rowwise softmax in LDS; normalized P written to HBM once.
//  - Pass 3: O = P V via WMMA, P read from LDS, V from L2 (32MB total KV).

#define AB 1
#define AH 16
#define AL 4096
#define AD 64
#define LP 4100              // LDS row stride (floats), padded to break bank conflicts
#define BLK_M 16             // query rows per workgroup
#define NWAVES 8             // 256 threads, wave32
#define RED_FLOATS (4 * 256) // cross-wave reduction scratch for pass 3

typedef __attribute__((ext_vector_type(2))) float v2f;
typedef __attribute__((ext_vector_type(8))) float v8f;

__device__ __forceinline__ v8f wmma_f32_4(v2f a, v2f b, v8f c) {
    // D = A(16x4,f32) * B(4x16,f32) + C(16x16,f32)
    return __builtin_amdgcn_wmma_f32_16x16x4_f32(
        /*neg_a=*/false, a, /*neg_b=*/false, b,
        /*c_mod=*/(short)0, c, /*reuse_a=*/false, /*reuse_b=*/false);
}

__global__ void __launch_bounds__(256)
attention_fused_kernel(const float* __restrict__ Q,
                       const float* __restrict__ K,
                       const float* __restrict__ V,
                       float* __restrict__ Out,
                       float* __restrict__ Score) {
    extern __shared__ float smem[];
    float* S   = smem;                          // [BLK_M][LP] score slab
    float* red = smem + (size_t)BLK_M * LP;     // [4][256] pass-3 partials

    const int head = blockIdx.y;
    const int qblk = blockIdx.x;
    const int tid  = threadIdx.x;
    const int wave = tid >> 5;
    const int lane = tid & 31;
    const int m    = lane & 15;  // M (or N) index within half-wave
    const int hi   = lane >> 4;  // half-wave selector (K phase per ISA A/B layout)

    const float* Qh = Q + ((size_t)head * AL + (size_t)qblk * BLK_M) * AD;
    const float* Kh = K + (size_t)head * AL * AD;
    const float* Vh = V + (size_t)head * AL * AD;

    const float scale = 0.125f; // 1/sqrt(64)

    // ---- Preload A operand: Q block (16x64), pre-scaled, per ISA 16x4 layout.
    // Lane (m,hi), k-step kq: a.x = Qs[m][4kq+2hi], a.y = Qs[m][4kq+2hi+1]
    v2f aq[16];
#pragma unroll
    for (int kq = 0; kq < 16; ++kq) {
        const int kk = 4 * kq + 2 * hi;
        aq[kq].x = Qh[m * AD + kk]     * scale;
        aq[kq].y = Qh[m * AD + kk + 1] * scale;
    }

    // ---- Pass 1: S tiles (16x16) across the 4096-wide row slab ------------
    // 256 N-tiles, round-robin over 8 waves.
    for (int t = wave; t < AL / 16; t += NWAVES) {
        const int n0 = t * 16;
        // B operand: B[k][n] = K[n0+n][k]; lane holds rows K=2hi, 2hi+1
        const float* Kt = Kh + (size_t)(n0 + m) * AD;
        v8f c0 = {};
        v8f c1 = {};
#pragma unroll
        for (int kq = 0; kq < 16; kq += 2) {
            const int kk0 = 4 * kq + 2 * hi;
            v2f b0; b0.x = Kt[kk0];     b0.y = Kt[kk0 + 1];
            c0 = wmma_f32_4(aq[kq], b0, c0);
            const int kk1 = kk0 + 4;
            v2f b1; b1.x = Kt[kk1];     b1.y = Kt[kk1 + 1];
            c1 = wmma_f32_4(aq[kq + 1], b1, c1);
        }
        const v8f c = c0 + c1;
        // C layout: v_r -> row (r + 8*hi), col (n0 + m)
#pragma unroll
        for (int r = 0; r < 8; ++r) {
            S[(size_t)(r + 8 * hi) * LP + n0 + m] = c[r];
        }
    }
    __syncthreads();

    // ---- Pass 2: rowwise softmax in LDS; write normalized scores to HBM ---
    // 16 rows over 8 waves: 2 rows per wave. Coalesced HBM store.
#pragma unroll
    for (int rr = 0; rr < 2; ++rr) {
        const int row = wave * 2 + rr;
        float* Srow = S + (size_t)row * LP;

        float mx = -3.402823466e+38f;
        for (int i = lane; i < AL; i += 32) mx = fmaxf(mx, Srow[i]);
#pragma unroll
        for (int off = 16; off > 0; off >>= 1)
            mx = fmaxf(mx, __shfl_xor(mx, off, 32));

        float sum = 0.0f;
        for (int i = lane; i < AL; i += 32) {
            const float e = __expf(Srow[i] - mx);
            Srow[i] = e;
            sum += e;
        }
#pragma unroll
        for (int off = 16; off > 0; off >>= 1)
            sum += __shfl_xor(sum, off, 32);
        const float inv = 1.0f / sum;

        float* dst = Score + ((size_t)head * AL + (size_t)qblk * BLK_M + row) * AL;
        for (int i = lane; i < AL; i += 32) {
            const float p = Srow[i] * inv;
            Srow[i] = p;      // keep normalized P in LDS for pass 3
            dst[i]  = p;      // the single mandatory HBM write of the score
        }
    }
    __syncthreads();

    // ---- Pass 3: O (16x64) = P (16x4096, LDS) x V (4096x64, L2) -----------
    // wave -> (ntile = wave&3 selects 16 output cols, khalf splits K range)
    const int ntile = wave & 3;
    const int khalf = wave >> 2;
    const float* Vn = Vh + ntile * 16 + m; // V[k][ntile*16 + m] column walker

    v8f c0 = {};
    v8f c1 = {};
    const int kbase = khalf * (AL / 2);
#pragma unroll 2
    for (int k0 = kbase; k0 < kbase + AL / 2; k0 += 8) {
        const int kk0 = k0 + 2 * hi;
        v2f a0; a0.x = S[(size_t)m * LP + kk0];   a0.y = S[(size_t)m * LP + kk0 + 1];
        v2f b0; b0.x = Vn[(size_t)kk0 * AD];      b0.y = Vn[(size_t)(kk0 + 1) * AD];
        c0 = wmma_f32_4(a0, b0, c0);

        const int kk1 = kk0 + 4;
        v2f a1; a1.x = S[(size_t)m * LP + kk1];   a1.y = S[(size_t)m * LP + kk1 + 1];
        v2f b1; b1.x = Vn[(size_t)kk1 * AD];      b1.y = Vn[(size_t)(kk1 + 1) * AD];
        c1 = wmma_f32_4(a1, b1, c1);
    }
    const v8f c = c0 + c1;

    // Cross-wave reduce the two K-halves, then write O.
    if (khalf == 1) {
#pragma unroll
        for (int r = 0; r < 8; ++r)
            red[ntile * 256 + r * 32 + lane] = c[r];
    }
    __syncthreads();
    if (khalf == 0) {
        float* Ob = Out + ((size_t)head * AL + (size_t)qblk * BLK_M) * AD;
#pragma unroll
        for (int r = 0; r < 8; ++r) {
            const float val = c[r] + red[ntile * 256 + r * 32 + lane];
            Ob[(size_t)(r + 8 * hi) * AD + ntile * 16 + m] = val;
        }
    }
}

extern "C" void kernel_launch(void* const* d_in, const int* in_sizes, int n_in,
                              void* d_out, int out_size, void* d_ws, size_t ws_size,
                              hipStream_t stream) {
    (void)in_sizes; (void)n_in; (void)out_size; (void)d_ws; (void)ws_size;

    const float* q = (const float*)d_in[0];
    const float* k = (const float*)d_in[1];
    const float* v = (const float*)d_in[2];

    float* out   = (float*)d_out;                                  // [1,16,4096,64]
    float* score = out + (size_t)AB * AH * AL * AD;                // [1,16,4096,4096]

    const dim3 grid(AL / BLK_M, AH);   // 256 x 16 workgroups
    const dim3 block(32 * NWAVES);     // 8 waves (wave32)
    const size_t shmem = ((size_t)BLK_M * LP + RED_FLOATS) * sizeof(float); // ~260 KB

    attention_fused_kernel<<<grid, block, shmem, stream>>>(q, k, v, out, score);
}